// BackProjection_75170517614722
// MI455X (gfx1250) — compile-verified
//
#include <hip/hip_runtime.h>
#include <hip/hip_bf16.h>

#ifndef __has_builtin
#define __has_builtin(x) 0
#endif

// ---- gfx1250 async global<->LDS path (ASYNCcnt). Probe result from round 1:
// builtins exist and take generic `int*` args: (int* gaddr, int* ldsaddr, imm, imm)
#if __has_builtin(__builtin_amdgcn_global_load_async_to_lds_b32)
#define HAVE_ASYNC_LOAD 1
#endif
#if __has_builtin(__builtin_amdgcn_global_store_async_from_lds_b32)
#define HAVE_ASYNC_STORE 1
#endif
#if __has_builtin(__builtin_amdgcn_s_wait_asynccnt)
#define WAIT_ASYNCCNT0() __builtin_amdgcn_s_wait_asynccnt(0)
#else
#define WAIT_ASYNCCNT0() asm volatile("s_wait_asynccnt 0" ::: "memory")
#endif

static constexpr float VOX = 0.05f;

// ---------------------------------------------------------------------------
// Setup: fold intrinsic^-1, frustum extents, camera2frustum^-1 and padding
// into 8 scalars per batch:  [a0, a1, b2, fx, fy, cx, cy, bi]
// pc_x = a0 - VOX*x ; pc_y = a1 - VOX*y ; depth = pc_z = b2 + VOX*z
// ---------------------------------------------------------------------------
__global__ void bp_setup(const float* __restrict__ intr,
                         float* __restrict__ cst, int B) {
  int b = blockIdx.x * blockDim.x + threadIdx.x;
  if (b >= B) return;
  const float* M = intr + (size_t)b * 16;
  float fx = M[0], cx = M[2], fy = M[5], cy = M[6];

  float xm = 1e30f, xM = -1e30f, ym = 1e30f, yM = -1e30f;
  const float dd[2] = {0.41f, 6.0f};
#pragma unroll
  for (int i = 0; i < 2; ++i) {
    float d = dd[i];
    float x0 = d * (0.0f   - cx) / fx;   // K^-1 unprojected frustum corners
    float x1 = d * (320.0f - cx) / fx;
    float y0 = d * (0.0f   - cy) / fy;
    float y1 = d * (240.0f - cy) / fy;
    xm = fminf(xm, fminf(x0, x1)); xM = fmaxf(xM, fmaxf(x0, x1));
    ym = fminf(ym, fminf(y0, y1)); yM = fmaxf(yM, fmaxf(y0, y1));
  }
  float mins0 = xm / VOX,     maxs0 = xM / VOX;
  float mins1 = ym / VOX,     maxs1 = yM / VOX;
  float mins2 = 0.41f / VOX,  maxs2 = 6.0f / VOX;
  float pad0 = (256.0f - ceilf(maxs0 - mins0)) * 0.5f;
  float pad1 = (256.0f - ceilf(maxs1 - mins1)) * 0.5f;
  float pad2 = (256.0f - ceilf(maxs2 - mins2)) * 0.5f;

  float* o = cst + b * 8;
  o[0] = VOX * ((255.0f - pad0) + mins0);   // a0
  o[1] = VOX * ((255.0f - pad1) + mins1);   // a1
  o[2] = VOX * ((mins2 - pad2) - 1.0f);     // b2
  o[3] = fx; o[4] = fy; o[5] = cx; o[6] = cy;
  o[7] = (float)b;
}

// ---------------------------------------------------------------------------
// Main: one block per (x,y) column, thread = z.  Block-uniform terms end up
// in SGPRs; frustum-mask ints DMA'd to LDS asynchronously; 5-channel AoS
// output transposed through LDS and streamed out as coalesced b32 bursts.
// ---------------------------------------------------------------------------
__global__ __launch_bounds__(256) void bp_main(
    const float* __restrict__ cst,
    const int*   __restrict__ fm,    // frustum masks (one int per voxel)
    const int*   __restrict__ rm,    // room masks (one int per pixel)
    const int*   __restrict__ shp,   // [H, W]
    float* __restrict__ outK,        // B*G^3 kept flags (0/1)
    float* __restrict__ outM) {      // B*G^3*5 mapping
  __shared__ int   smask[256];
  __shared__ float smap[5 * 256];

  const int      tid = threadIdx.x;
  const unsigned bx  = blockIdx.x;          // x*256 + y
  const int      b   = blockIdx.y;
  const int x = (int)(bx >> 8);
  const int y = (int)(bx & 255u);

  const size_t vbase = ((size_t)b << 24) + ((size_t)bx << 8);

  // Kick off mask fetch (global -> LDS, ASYNCcnt) before the ALU work.
#if HAVE_ASYNC_LOAD
  __builtin_amdgcn_global_load_async_to_lds_b32(
      (int*)(fm + vbase + tid), &smask[tid], 0, 0);
#else
  smask[tid] = fm[vbase + tid];
#endif

  const float* c = cst + b * 8;              // uniform -> scalar loads
  const float a0 = c[0], a1 = c[1], b2 = c[2];
  const float fx = c[3], fy = c[4], cxv = c[5], cyv = c[6], bi = c[7];
  const int H = shp[0], W = shp[1];
  const float hF = (float)H, wF = (float)W;

  const float pc0 = a0 - VOX * (float)x;     // block-uniform
  const float pc1 = a1 - VOX * (float)y;     // block-uniform
  const float pc2 = b2 + VOX * (float)tid;   // per-lane depth
  const float dpx = fmaf(cxv, pc2, fx * pc0);
  const float dpy = fmaf(cyv, pc2, fy * pc1);
  const float px  = dpx / pc2;               // IEEE div, matches jnp
  const float py  = dpy / pc2;

#if HAVE_ASYNC_LOAD
  WAIT_ASYNCCNT0();
#endif
  const bool kept = (pc2 <= 6.0f) & (pc2 >= 0.41f) &
                    (px < wF) & (py < hF) & (smask[tid] != 0);

  const float zf = (float)tid;
  smap[tid * 5 + 0] = kept ? bi  : -1.0f;
  smap[tid * 5 + 1] = kept ? px  : -1.0f;
  smap[tid * 5 + 2] = kept ? py  : -1.0f;
  smap[tid * 5 + 3] = kept ? zf  : -1.0f;
  smap[tid * 5 + 4] = kept ? pc2 : -1.0f;

  // kept2 = all(mapping >= 0) & room_mask[floor(py), floor(px)]
  bool k2 = kept & (px >= 0.0f) & (py >= 0.0f);
  int ridx = 0;
  if (k2) ridx = (int)py * W + (int)px;      // in-bounds when k2
  k2 = k2 & (rm[(size_t)b * (size_t)H * (size_t)W + ridx] != 0);
  outK[vbase + tid] = k2 ? 1.0f : 0.0f;

  __syncthreads();                           // smap fully populated
  float* dst = outM + vbase * 5;             // 1280 contiguous floats/block
#if HAVE_ASYNC_STORE
#pragma unroll
  for (int i = 0; i < 5; ++i)
    __builtin_amdgcn_global_store_async_from_lds_b32(
        (int*)(dst + i * 256 + tid), (int*)&smap[i * 256 + tid], 0, 0);
  // s_endpgm performs an implicit wait-idle; LDS stays live until then.
#else
#pragma unroll
  for (int i = 0; i < 5; ++i)
    dst[i * 256 + tid] = smap[i * 256 + tid];
#endif
}

// ---------------------------------------------------------------------------
extern "C" void kernel_launch(void* const* d_in, const int* in_sizes, int n_in,
                              void* d_out, int out_size, void* d_ws, size_t ws_size,
                              hipStream_t stream) {
  const int*   shp  = (const int*)  d_in[0];
  const float* intr = (const float*)d_in[1];
  const int*   fm   = (const int*)  d_in[2];
  const int*   rm   = (const int*)  d_in[3];
  const int B = in_sizes[1] / 16;

  float* cst = (float*)d_ws;                         // 8 floats per batch
  bp_setup<<<dim3((B + 63) / 64), 64, 0, stream>>>(intr, cst, B);

  float* outK = (float*)d_out;
  float* outM = outK + (size_t)B * (1u << 24);       // kepts first, then maps
  dim3 grid(256 * 256, B);
  bp_main<<<grid, 256, 0, stream>>>(cst, fm, rm, shp, outK, outM);
  (void)n_in; (void)out_size; (void)ws_size;
}